// GraphTransformer_41755672051883
// MI455X (gfx1250) — compile-verified
//
#include <hip/hip_runtime.h>
#include <hip/hip_bf16.h>

// ---------------------------------------------------------------------------
// Graph-transformer forward for MI455X (gfx1250, wave32, WMMA).
//
//  * Projections hoisted from edges (E=640K) to nodes (N=50K):
//        embeds[rows] @ W == (embeds @ W)[rows]   (12.8x fewer GEMM FLOPs)
//  * GEMMs: V_WMMA_F32_16X16X4_F32 (exact fp32; workload is L2-bound so
//    there is no reason to quantize).
//  * CSR adjacency built once per call; aggregation + residual + LayerNorm
//    fused into one wave-per-node kernel => zero float atomics on the E x D
//    path (vs 164M in the naive atomic formulation).
//  * Q/K/V (76.8 MB) are resident in the 192 MB L2; next-edge V rows are
//    prefetched (global_prefetch_b8) through the edge-list indirection.
// ---------------------------------------------------------------------------

#define D_MODEL 128
#define NHEAD   8
#define DHEAD   16
#define NLAYER  2

typedef __attribute__((ext_vector_type(2))) float v2f;
typedef __attribute__((ext_vector_type(8))) float v8f;

// ---------------------------------------------------------------------------
// out[M x 128] = A[M x 128] @ W[128 x 128] (+ bias) (+ pos)
// blockDim = (32, 8): one wave per 16x16 tile, 32 chained WMMA issues (K=128).
// Fragment layouts per CDNA5 ISA 7.12.2 (32-bit, wave32).
// ---------------------------------------------------------------------------
__global__ __launch_bounds__(256)
void gemm_wmma_f32(const float* __restrict__ A,
                   const float* __restrict__ W,
                   const float* __restrict__ bias,   // nullable
                   const float* __restrict__ pos,    // nullable
                   float* __restrict__ out)
{
    const int lane  = threadIdx.x;          // 0..31
    const int ct    = threadIdx.y;          // column tile 0..7
    const int row0  = blockIdx.x * 16;
    const int col0  = ct * 16;

    const int mn    = lane & 15;            // M index for A, N index for B/C
    const int khalf = (lane >> 4) * 2;      // 0 for lanes 0-15, 2 for 16-31

    v8f acc = {};
    const float* arow = A + (size_t)(row0 + mn) * D_MODEL;

#pragma unroll 4
    for (int k0 = 0; k0 < D_MODEL; k0 += 4) {
        v2f a = *(const v2f*)(arow + k0 + khalf);
        v2f b;
        b.x = W[(size_t)(k0 + khalf    ) * D_MODEL + col0 + mn];
        b.y = W[(size_t)(k0 + khalf + 1) * D_MODEL + col0 + mn];
        acc = __builtin_amdgcn_wmma_f32_16x16x4_f32(
            false, a, false, b, (short)0, acc, false, false);
    }

    float add = 0.0f;
    if (bias) add += bias[col0 + mn];
    if (pos)  add += pos[col0 + mn];

    const int mbase = (lane >> 4) * 8;      // C/D layout: upper lanes hold M+8
#pragma unroll
    for (int r = 0; r < 8; ++r) {
        out[(size_t)(row0 + mbase + r) * D_MODEL + col0 + mn] = acc[r] + add;
    }
}

// ---------------------------------------------------------------------------
__global__ void fill_zero_f(float* __restrict__ p, int n)
{
    int i = blockIdx.x * blockDim.x + threadIdx.x;
    if (i < n) p[i] = 0.0f;
}

__global__ void fill_zero_i(int* __restrict__ p, int n)
{
    int i = blockIdx.x * blockDim.x + threadIdx.x;
    if (i < n) p[i] = 0;
}

// ---------------------------------------------------------------------------
// CSR build, stage 1: per-row degree count; the atomic's pre-op return is
// kept as this edge's slot (rank) within its row.
// ---------------------------------------------------------------------------
__global__ void count_rank(const int* __restrict__ rows,
                           int* __restrict__ count,
                           int* __restrict__ rank,
                           int nE)
{
    int e = blockIdx.x * blockDim.x + threadIdx.x;
    if (e >= nE) return;
    rank[e] = atomicAdd(&count[rows[e]], 1);
}

// ---------------------------------------------------------------------------
// CSR build, stage 2: exclusive scan of degrees -> rowStart[0..nN].
// Single workgroup (256 threads), chunked; ~nN/256 serial adds per thread.
// ---------------------------------------------------------------------------
__global__ __launch_bounds__(256)
void scan_rows(const int* __restrict__ count,
               int* __restrict__ rowStart,
               int nN)
{
    __shared__ int partial[256];
    int t = threadIdx.x;
    int chunk = (nN + 255) / 256;
    int begin = t * chunk;
    int end   = begin + chunk < nN ? begin + chunk : nN;

    int s = 0;
    for (int i = begin; i < end; ++i) s += count[i];
    partial[t] = s;
    __syncthreads();

    if (t == 0) {
        int acc = 0;
        for (int i = 0; i < 256; ++i) { int v = partial[i]; partial[i] = acc; acc += v; }
        rowStart[nN] = acc;            // == nE
    }
    __syncthreads();

    int acc = partial[t];
    for (int i = begin; i < end; ++i) { rowStart[i] = acc; acc += count[i]; }
}

// ---------------------------------------------------------------------------
// CSR build, stage 3: scatter edge ids into their row segments.
// ---------------------------------------------------------------------------
__global__ void scatter_edges(const int* __restrict__ rows,
                              const int* __restrict__ rank,
                              const int* __restrict__ rowStart,
                              int* __restrict__ edgeIds,
                              int nE)
{
    int e = blockIdx.x * blockDim.x + threadIdx.x;
    if (e >= nE) return;
    edgeIds[rowStart[rows[e]] + rank[e]] = e;
}

// ---------------------------------------------------------------------------
// Per (edge, head): att = clip(q . k, -10, 10); expAtt stored; denom += exp.
// Q/K gathers hit the 192 MB L2 (25.6 MB each).
// ---------------------------------------------------------------------------
__global__ void edge_att(const float* __restrict__ Q,
                         const float* __restrict__ K,
                         const int*   __restrict__ rows,
                         const int*   __restrict__ cols,
                         float* __restrict__ expAtt,
                         float* __restrict__ denom,
                         int nE)
{
    int t = blockIdx.x * blockDim.x + threadIdx.x;
    if (t >= nE * NHEAD) return;
    int e = t >> 3;
    int h = t & 7;
    int r = rows[e];
    int c = cols[e];

    const float4* q = (const float4*)(Q + (size_t)r * D_MODEL + h * DHEAD);
    const float4* k = (const float4*)(K + (size_t)c * D_MODEL + h * DHEAD);

    float s = 0.0f;
#pragma unroll
    for (int j = 0; j < 4; ++j) {
        float4 a = q[j], b = k[j];
        s += a.x * b.x + a.y * b.y + a.z * b.z + a.w * b.w;
    }
    s = fminf(10.0f, fmaxf(-10.0f, s));
    float ex = expf(s);
    expAtt[t] = ex;
    atomicAdd(&denom[r * NHEAD + h], ex);   // 1.6 MB, L2-resident
}

// ---------------------------------------------------------------------------
// Fused softmax-weighted aggregation + residual + LayerNorm.
// One wave32 per node: lane owns 4 feature dims (all within head lane>>2).
// Walks the node's CSR edge list; V rows gathered from L2; next edge's V row
// is prefetched through the indirection (global_prefetch_b8).
// No float atomics; per-node accumulation order fixed by the edge list.
// ---------------------------------------------------------------------------
__global__ __launch_bounds__(256)
void node_agg_ln(const float* __restrict__ V,
                 const int*   __restrict__ cols,
                 const int*   __restrict__ edgeIds,
                 const int*   __restrict__ rowStart,
                 const float* __restrict__ expAtt,
                 const float* __restrict__ denom,
                 float* __restrict__ embeds,
                 const float* __restrict__ g,
                 const float* __restrict__ bta,
                 int nN)
{
    int node = (blockIdx.x * blockDim.x + threadIdx.x) >> 5;
    int lane = threadIdx.x & 31;
    if (node >= nN) return;

    const int h = lane >> 2;          // head for this lane's 4 dims
    const int d = lane * 4;
    const float invDen = 1.0f / (denom[node * NHEAD + h] + 1e-8f);

    const int start = rowStart[node];
    const int end   = rowStart[node + 1];

    float ax = 0.0f, ay = 0.0f, az = 0.0f, aw = 0.0f;
    for (int idx = start; idx < end; ++idx) {
        int e = edgeIds[idx];
        int c = cols[e];
        if (idx + 1 < end) {  // prefetch next edge's V row (hides L2 latency)
            __builtin_prefetch(V + (size_t)cols[edgeIds[idx + 1]] * D_MODEL + d, 0, 0);
        }
        float w = expAtt[(size_t)e * NHEAD + h] * invDen;
        float4 v = *(const float4*)(V + (size_t)c * D_MODEL + d);
        ax += w * v.x; ay += w * v.y; az += w * v.z; aw += w * v.w;
    }

    // Residual
    size_t base = (size_t)node * D_MODEL + d;
    float4 eb = *(const float4*)(embeds + base);
    float xx = ax + eb.x, xy = ay + eb.y, xz = az + eb.z, xw = aw + eb.w;

    // LayerNorm over 128 dims (wave32 butterfly reductions)
    float s = xx + xy + xz + xw;
#pragma unroll
    for (int off = 16; off >= 1; off >>= 1) s += __shfl_xor(s, off, 32);
    float mu = s * (1.0f / D_MODEL);

    float dx = xx - mu, dy = xy - mu, dz = xz - mu, dw = xw - mu;
    float s2 = dx * dx + dy * dy + dz * dz + dw * dw;
#pragma unroll
    for (int off = 16; off >= 1; off >>= 1) s2 += __shfl_xor(s2, off, 32);
    float rs = rsqrtf(s2 * (1.0f / D_MODEL) + 1e-6f);

    float4 o;
    o.x = dx * rs * g[d + 0] + bta[d + 0];
    o.y = dy * rs * g[d + 1] + bta[d + 1];
    o.z = dz * rs * g[d + 2] + bta[d + 2];
    o.w = dw * rs * g[d + 3] + bta[d + 3];
    *(float4*)(embeds + base) = o;
}

// ---------------------------------------------------------------------------
extern "C" void kernel_launch(void* const* d_in, const int* in_sizes, int n_in,
                              void* d_out, int out_size, void* d_ws, size_t ws_size,
                              hipStream_t stream)
{
    const float* nodef = (const float*)d_in[0];    // [N, 128]
    const int*   eidx  = (const int*)  d_in[1];    // [2, E]
    const float* WPw   = (const float*)d_in[2];    // [128, 128]
    const float* WPb   = (const float*)d_in[3];    // [128]
    const float* Wpos  = (const float*)d_in[4];    // [1, 128]
    const float* qT    = (const float*)d_in[5];    // [L, 128, 128]
    const float* kT    = (const float*)d_in[6];
    const float* vT    = (const float*)d_in[7];
    const float* lng   = (const float*)d_in[8];    // [L, 128]
    const float* lnb   = (const float*)d_in[9];
    const float* invW  = (const float*)d_in[10];   // [128, 128]
    const float* invb  = (const float*)d_in[11];   // [128]
    float* out = (float*)d_out;

    const int nN = in_sizes[0] / D_MODEL;          // 50000 (multiple of 16)
    const int nE = in_sizes[1] / 2;                // 640000
    const int* rows = eidx;
    const int* cols = eidx + nE;

    // ---- Workspace layout (~130 MB) ----
    char* wsb = (char*)d_ws;
    size_t off = 0;
    auto carveF = [&](size_t n) { float* p = (float*)(wsb + off); off += n * sizeof(float); return p; };
    auto carveI = [&](size_t n) { int*   p = (int*)  (wsb + off); off += ((n + 3) & ~(size_t)3) * sizeof(int); return p; };

    float* embeds  = carveF((size_t)nN * D_MODEL);
    float* Q       = carveF((size_t)nN * D_MODEL);
    float* Kb      = carveF((size_t)nN * D_MODEL);
    float* Vb      = carveF((size_t)nN * D_MODEL);
    float* expAtt  = carveF((size_t)nE * NHEAD);
    float* denom   = carveF((size_t)nN * NHEAD);
    int*   count   = carveI(nN);
    int*   rank    = carveI(nE);
    int*   rowStart= carveI(nN + 1);
    int*   edgeIds = carveI(nE);

    const dim3 gemmBlk(32, 8);
    const int  gemmGrid = nN / 16;

    // ---- CSR adjacency: built once per call (edge_index is layer-invariant)
    fill_zero_i<<<(nN + 255) / 256, 256, 0, stream>>>(count, nN);
    count_rank<<<(nE + 255) / 256, 256, 0, stream>>>(rows, count, rank, nE);
    scan_rows<<<1, 256, 0, stream>>>(count, rowStart, nN);
    scatter_edges<<<(nE + 255) / 256, 256, 0, stream>>>(rows, rank, rowStart, edgeIds, nE);

    // ---- 0) embeds = X @ W_P + b + W_pos
    gemm_wmma_f32<<<gemmGrid, gemmBlk, 0, stream>>>(nodef, WPw, WPb, Wpos, embeds);

    for (int l = 0; l < NLAYER; ++l) {
        const float* qW = qT + (size_t)l * D_MODEL * D_MODEL;
        const float* kW = kT + (size_t)l * D_MODEL * D_MODEL;
        const float* vW = vT + (size_t)l * D_MODEL * D_MODEL;

        // Per-node projections (hoisted out of the edge gather)
        gemm_wmma_f32<<<gemmGrid, gemmBlk, 0, stream>>>(embeds, qW, nullptr, nullptr, Q);
        gemm_wmma_f32<<<gemmGrid, gemmBlk, 0, stream>>>(embeds, kW, nullptr, nullptr, Kb);
        gemm_wmma_f32<<<gemmGrid, gemmBlk, 0, stream>>>(embeds, vW, nullptr, nullptr, Vb);

        // Softmax denominators (zeroed every call; harness does not re-poison)
        fill_zero_f<<<(nN * NHEAD + 255) / 256, 256, 0, stream>>>(denom, nN * NHEAD);
        edge_att<<<(nE * NHEAD + 255) / 256, 256, 0, stream>>>(Q, Kb, rows, cols,
                                                               expAtt, denom, nE);

        // Fused CSR aggregation + residual + LayerNorm (no float atomics)
        node_agg_ln<<<(nN + 7) / 8, 256, 0, stream>>>(Vb, cols, edgeIds, rowStart,
                                                      expAtt, denom, embeds,
                                                      lng + (size_t)l * D_MODEL,
                                                      lnb + (size_t)l * D_MODEL, nN);
    }

    // ---- Final projection -> d_out
    gemm_wmma_f32<<<gemmGrid, gemmBlk, 0, stream>>>(embeds, invW, invb, nullptr, out);
}